// GCNLayer_14362370638268
// MI455X (gfx1250) — compile-verified
//
#include <hip/hip_runtime.h>
#include <hip/hip_bf16.h>

typedef __attribute__((ext_vector_type(2))) float v2f;
typedef __attribute__((ext_vector_type(8))) float v8f;

#define N_NODES 50000
#define N_EDGES 800000
#define IN_F 256
#define OUT_F 64
#define W_STRIDE 160   // floats per paired-K row: 128 payload + 32 swizzle pad

// ---------------------------------------------------------------------------
// Kernel 1: hw[n, :] = (h[n, :] @ W) * norm[n]
// One wave computes a 16(M) x 64(N) tile via V_WMMA_F32_16X16X4_F32.
// W staged raw into LDS with GLOBAL_LOAD_ASYNC_TO_LDS_B128 (ASYNCcnt),
// then repacked into a paired-K, bank-swizzled layout so every B fragment
// is a single conflict-free ds_load_b64.
// ---------------------------------------------------------------------------
__global__ __launch_bounds__(256) void gcn_gemm_norm(
    const float* __restrict__ h, const float* __restrict__ norm,
    const float* __restrict__ W, float* __restrict__ hw)
{
    __shared__ float Wraw[IN_F * OUT_F];          // 64 KB raw copy
    __shared__ float Wp[(IN_F / 2) * W_STRIDE];   // 80 KB paired-K layout

    // --- Stage 1: async copy W (64 KB) global -> LDS -----------------------
    {
        const int t = threadIdx.x;
        #pragma unroll
        for (int i = 0; i < (IN_F * OUT_F / 4) / 256; ++i) {   // 16 x float4
            const int v4 = i * 256 + t;
            unsigned int       loff = (unsigned int)(size_t)(const void*)&Wraw[v4 * 4];
            unsigned long long ga   = (unsigned long long)(size_t)(const void*)(W + v4 * 4);
            asm volatile("global_load_async_to_lds_b128 %0, %1, off"
                         :: "v"(loff), "v"(ga) : "memory");
        }
        asm volatile("s_wait_asynccnt 0" ::: "memory");
    }
    __syncthreads();

    // --- Stage 2: repack to paired-K: Wp[(k/2)*160 + 2n + (k&1)] = W[k][n] -
    for (int f = threadIdx.x; f < IN_F * OUT_F; f += 256) {
        const int k = f >> 6;
        const int n = f & 63;
        Wp[(k >> 1) * W_STRIDE + 2 * n + (k & 1)] = Wraw[f];
    }
    __syncthreads();

    const int wave = threadIdx.x >> 5;
    const int lane = threadIdx.x & 31;
    const int tile = blockIdx.x * 8 + wave;        // 16-row M tile
    if (tile >= N_NODES / 16) return;              // 50000 % 16 == 0

    const int row0 = tile * 16;
    const int half = lane >> 4;                    // K pair selector
    const int l16  = lane & 15;

    v8f acc0 = {}, acc1 = {}, acc2 = {}, acc3 = {};

    const float* hrow  = h + (size_t)(row0 + l16) * IN_F;
    const float* wbase = Wp + half * W_STRIDE + 2 * l16;

    #pragma unroll 4
    for (int k2 = 0; k2 < IN_F / 2; k2 += 2) {     // k2 = k/2, K step of 4
        // A fragment: h[row0+l16][4*? ...] -> lane holds K = 2*k2+2*half, +1
        v2f a = *(const v2f*)(hrow + 2 * k2 + 2 * half);

        // B fragments: one contiguous b64 per N tile, conflict-free banks
        const float* wp = wbase + k2 * W_STRIDE;
        v2f b0 = *(const v2f*)(wp +  0);
        v2f b1 = *(const v2f*)(wp + 32);
        v2f b2 = *(const v2f*)(wp + 64);
        v2f b3 = *(const v2f*)(wp + 96);

        acc0 = __builtin_amdgcn_wmma_f32_16x16x4_f32(false, a, false, b0, (short)0, acc0, false, false);
        acc1 = __builtin_amdgcn_wmma_f32_16x16x4_f32(false, a, false, b1, (short)0, acc1, false, false);
        acc2 = __builtin_amdgcn_wmma_f32_16x16x4_f32(false, a, false, b2, (short)0, acc2, false, false);
        acc3 = __builtin_amdgcn_wmma_f32_16x16x4_f32(false, a, false, b3, (short)0, acc3, false, false);
    }

    // D element r maps to row = row0 + r + 8*half, col = n0 + l16.
    const float* np = norm + row0 + 8 * half;
    float*       op = hw + (size_t)(row0 + 8 * half) * OUT_F + l16;
    #pragma unroll
    for (int r = 0; r < 8; ++r) {
        const float nv = np[r];
        op[(size_t)r * OUT_F +  0] = acc0[r] * nv;
        op[(size_t)r * OUT_F + 16] = acc1[r] * nv;
        op[(size_t)r * OUT_F + 32] = acc2[r] * nv;
        op[(size_t)r * OUT_F + 48] = acc3[r] * nv;
    }
}

// ---------------------------------------------------------------------------
// Kernel 2: zero the aggregation buffer (deterministic per call).
// ---------------------------------------------------------------------------
__global__ __launch_bounds__(256) void gcn_zero(float4* __restrict__ agg4, int n4)
{
    int t = blockIdx.x * 256 + threadIdx.x;
    if (t < n4) agg4[t] = make_float4(0.f, 0.f, 0.f, 0.f);
}

// ---------------------------------------------------------------------------
// Kernel 3: edge-parallel scatter-add. One thread = (edge, 4-float chunk).
// hw/agg are 12.8 MB each -> resident in 192 MB L2; f32 atomics, no return.
// ---------------------------------------------------------------------------
__global__ __launch_bounds__(256) void gcn_scatter(
    const float* __restrict__ hw, const int* __restrict__ src,
    const int* __restrict__ dst, float* __restrict__ agg)
{
    int t = blockIdx.x * 256 + threadIdx.x;
    if (t >= N_EDGES * 16) return;
    const int e = t >> 4;
    const int c = (t & 15) * 4;
    const int s = src[e];
    const int d = dst[e];
    const float4 v = *(const float4*)(hw + (size_t)s * OUT_F + c);
    float* o = agg + (size_t)d * OUT_F + c;
    (void)__hip_atomic_fetch_add(o + 0, v.x, __ATOMIC_RELAXED, __HIP_MEMORY_SCOPE_AGENT);
    (void)__hip_atomic_fetch_add(o + 1, v.y, __ATOMIC_RELAXED, __HIP_MEMORY_SCOPE_AGENT);
    (void)__hip_atomic_fetch_add(o + 2, v.z, __ATOMIC_RELAXED, __HIP_MEMORY_SCOPE_AGENT);
    (void)__hip_atomic_fetch_add(o + 3, v.w, __ATOMIC_RELAXED, __HIP_MEMORY_SCOPE_AGENT);
}

// ---------------------------------------------------------------------------
// Kernel 4: out = relu(agg * norm + bias)
// ---------------------------------------------------------------------------
__global__ __launch_bounds__(256) void gcn_finalize(
    const float* __restrict__ agg, const float* __restrict__ norm,
    const float* __restrict__ bias, float* __restrict__ out)
{
    int t = blockIdx.x * 256 + threadIdx.x;
    if (t >= N_NODES * OUT_F) return;
    const int row = t >> 6;      // /64
    const int col = t & 63;
    float v = agg[t] * norm[row] + bias[col];
    out[t] = v > 0.f ? v : 0.f;
}

extern "C" void kernel_launch(void* const* d_in, const int* in_sizes, int n_in,
                              void* d_out, int out_size, void* d_ws, size_t ws_size,
                              hipStream_t stream)
{
    const float* h    = (const float*)d_in[0];
    const float* norm = (const float*)d_in[1];
    const float* W    = (const float*)d_in[2];
    const float* bias = (const float*)d_in[3];
    const int*   src  = (const int*)d_in[4];
    const int*   dst  = (const int*)d_in[5];
    float*       out  = (float*)d_out;

    float* hw  = (float*)d_ws;                       // 50000*64 f32
    float* agg = hw + (size_t)N_NODES * OUT_F;       // 50000*64 f32

    // GEMM + pre-norm: 3125 M-tiles, 8 waves (tiles) per block
    const int n_tiles = N_NODES / 16;                // 3125
    gcn_gemm_norm<<<(n_tiles + 7) / 8, 256, 0, stream>>>(h, norm, W, hw);

    // Zero agg
    const int n4 = (N_NODES * OUT_F) / 4;            // 800000 float4
    gcn_zero<<<(n4 + 255) / 256, 256, 0, stream>>>((float4*)agg, n4);

    // Edge scatter-add: 800000 edges * 16 chunks
    const long long n_sc = (long long)N_EDGES * 16;
    gcn_scatter<<<(int)((n_sc + 255) / 256), 256, 0, stream>>>(hw, src, dst, agg);

    // Finalize
    const int n_out = N_NODES * OUT_F;
    gcn_finalize<<<(n_out + 255) / 256, 256, 0, stream>>>(agg, norm, bias, out);
}